// ExactWeightedDTM_51359218925740
// MI455X (gfx1250) — compile-verified
//
#include <hip/hip_runtime.h>
#include <hip/hip_bf16.h>
#include <stdint.h>

// ---------------------------------------------------------------------------
// ExactWeightedDTM on gfx1250 (MI455X).
// B=4, H=W=64, N=4096, R=2 (dists_pow == integer squared distance d2),
// M0=0.01.  Result depends only on mass consumed per distinct d2 value, so a
// distance-ordered offset table (counting sort over d2 in [0,7938]) replaces
// the reference's full N x N argsort.
//
// Workspace (ints): [0, NDIST) counting-sort hist/cursors,
//                   [NDIST, NDIST+NOFFS) sorted offset table.
// Needs (7939 + 16129) * 4 = 96,272 bytes of d_ws.
// ---------------------------------------------------------------------------

#define NDIST 7939          // d2 in [0, 63^2 + 63^2] = [0, 7938]
#define NOFFS (127 * 127)   // all (dy,dx) in [-63,63]^2

typedef __attribute__((ext_vector_type(2))) float v2f;
typedef __attribute__((ext_vector_type(8))) float v8f;
typedef __attribute__((ext_vector_type(4))) unsigned int u32x4;
typedef __attribute__((ext_vector_type(4))) int i32x4;
typedef __attribute__((ext_vector_type(8))) int i32x8;

typedef __attribute__((address_space(1))) i32x4 gv4i;   // global v4i
typedef __attribute__((address_space(3))) i32x4 lv4i;   // LDS v4i

#define DTM_STR2(x) #x
#define DTM_STR(x) DTM_STR2(x)
#pragma message("CDNA5 probe: __clang_major__ = " DTM_STR(__clang_major__))

#if __has_builtin(__builtin_amdgcn_tensor_load_to_lds) && __has_builtin(__builtin_amdgcn_s_wait_tensorcnt)
#define HAVE_TDM 1
#warning "CDNA5 probe: tensor_load_to_lds builtin AVAILABLE (TDM path compiled in)"
#else
#define HAVE_TDM 0
#warning "CDNA5 probe: tensor_load_to_lds builtin MISSING (TDM path compiled out)"
#endif

#if __has_builtin(__builtin_amdgcn_global_load_async_to_lds_b128) && __has_builtin(__builtin_amdgcn_s_wait_asynccnt)
#define HAVE_ASYNC 1
#warning "CDNA5 probe: global_load_async_to_lds_b128 builtin AVAILABLE (async copy path)"
#else
#define HAVE_ASYNC 0
#warning "CDNA5 probe: global_load_async_to_lds_b128 builtin MISSING (plain copy path)"
#endif

#if __has_builtin(__builtin_amdgcn_wmma_f32_16x16x4_f32)
#define HAVE_WMMA4 1
#else
#define HAVE_WMMA4 0
#warning "CDNA5 probe: wmma_f32_16x16x4_f32 builtin MISSING (shfl reduction fallback)"
#endif

// --------------------------- setup kernels ---------------------------------

__global__ void dtm_hist_init(int* hist) {
    int i = blockIdx.x * blockDim.x + threadIdx.x;
    if (i < NDIST) hist[i] = 0;
}

__global__ void dtm_hist_count(int* hist) {
    int idx = blockIdx.x * blockDim.x + threadIdx.x;
    if (idx >= NOFFS) return;
    int dy = idx / 127 - 63;
    int dx = idx % 127 - 63;
    int d2 = dy * dy + dx * dx;
    atomicAdd(&hist[d2], 1);
}

// Serial scan + serial scatter on one thread: ~24K trivial iterations, and it
// makes the within-d2 ordering (hence downstream fp summation order) fully
// deterministic across calls (harness revalidates bit-identical output).
__global__ void dtm_scan_scatter(int* hist, int* table) {
    if (threadIdx.x != 0 || blockIdx.x != 0) return;
    int run = 0;
    for (int d = 0; d < NDIST; ++d) {
        int c = hist[d];
        hist[d] = run;          // exclusive prefix -> cursor
        run += c;
    }
    for (int idx = 0; idx < NOFFS; ++idx) {
        int dy = idx / 127 - 63;
        int dx = idx % 127 - 63;
        int d2 = dy * dy + dx * dx;
        int pos = hist[d2]++;
        table[pos] = (d2 << 16) | ((dy + 63) << 8) | (dx + 63);
    }
}

// ----------------------------- main kernel ---------------------------------
// grid = 64 blocks (16 per batch image, 4 rows of 64 pixels each),
// block = 256 threads (8 wave32s). Whole 64x64 image lives in LDS (16 KB of
// the WGP's 320 KB).

__global__ __launch_bounds__(256) void dtm_main(const float* __restrict__ images,
                                                float* __restrict__ out,
                                                const int* __restrict__ table) {
    __shared__ __align__(16) float img[4096];

    const int b  = blockIdx.x >> 4;   // batch
    const int rb = blockIdx.x & 15;   // 4-row group within image
    const float* src = images + b * 4096;

#if HAVE_TDM
    // Tensor Data Mover: DMA the full 64x64 f32 image into LDS as a 1-D tile.
    // Issued unconditionally (TDM ignores EXEC; per-wave op); every wave then
    // waits on its own TENSORcnt before the barrier.
    {
        uint64_t ga  = (uint64_t)(uintptr_t)src;
        uint32_t lds = (uint32_t)(uintptr_t)(&img[0]);   // low 32 bits = LDS byte offset
        // D# group 0: count=1 (valid user desc), lds_addr, global_addr[56:0], type=2
        u32x4 g0 = { 1u,
                     lds,
                     (uint32_t)(ga & 0xFFFFFFFFu),
                     (uint32_t)((ga >> 32) & 0x1FFFFFFu) | (2u << 30) };
        // D# group 1: data_size=2 (4B), tensor_dim0=4096, tensor_dim1=1,
        //             tile_dim0=4096, tile_dim1/2=0, dim0 stride = 4096.
        i32x8 g1 = { (int)(2u << 16),
                     (int)(4096u << 16),
                     (int)(1u << 16),
                     (int)(4096u << 16),
                     0,
                     4096,
                     0, 0 };
        i32x4 zz = { 0, 0, 0, 0 };
#if __clang_major__ >= 23
        i32x8 z8 = { 0, 0, 0, 0, 0, 0, 0, 0 };
        __builtin_amdgcn_tensor_load_to_lds(g0, g1, zz, zz, z8, 0);
#else
        __builtin_amdgcn_tensor_load_to_lds(g0, g1, zz, zz, 0);
#endif
        __builtin_amdgcn_s_wait_tensorcnt(0);
    }
#endif
    __syncthreads();

    // Fill LDS with the image. Prefer the gfx1250 async-DMA path
    // (GLOBAL_LOAD_ASYNC_TO_LDS_B128, tracked by ASYNCcnt); it also rewrites
    // the same bytes the experimental TDM descriptor targeted, so output
    // correctness never depends on my D# bit-packing.
#if HAVE_ASYNC
    for (int i = threadIdx.x; i < 1024; i += 256) {
        gv4i* gsrc = (gv4i*)((uint64_t)(uintptr_t)src + (uint64_t)(i * 16));
        lv4i* ldst = (lv4i*)((uint32_t)(uintptr_t)(&img[0]) + (uint32_t)(i * 16));
        __builtin_amdgcn_global_load_async_to_lds_b128(gsrc, ldst, 0, 0);
    }
    __builtin_amdgcn_s_wait_asynccnt(0);
#else
    for (int i = threadIdx.x; i < 1024; i += 256)
        ((float4*)img)[i] = ((const float4*)src)[i];
#endif
    __syncthreads();

    // ---- total_mass = sum(img) via V_WMMA_F32_16X16X4_F32 -----------------
    // B matrix = all ones => D[m][n] = sum_k A[m][k] (+C). Since B is ones and
    // we sum every row of D, ANY bijection of the 64 chunk values onto A's
    // lane/VGPR slots is valid. All 8 waves redundantly compute the identical
    // full reduction (EXEC all-ones for WMMA, bit-deterministic, no atomics).
    const int lane = threadIdx.x & 31;
    float total;
#if HAVE_WMMA4
    {
        v8f acc = {};
        v2f bm = { 1.0f, 1.0f };
        for (int c = 0; c < 64; ++c) {
            v2f a = { img[c * 64 + lane * 2], img[c * 64 + lane * 2 + 1] };
            acc = __builtin_amdgcn_wmma_f32_16x16x4_f32(
                false, a, false, bm, (short)0, acc, false, false);
        }
        float s = acc[0] + acc[1] + acc[2] + acc[3] +
                  acc[4] + acc[5] + acc[6] + acc[7];
        // rows 0..7 accumulate in lane 0, rows 8..15 in lane 16 (N=0 column)
        total = __shfl(s, 0, 32) + __shfl(s, 16, 32);
    }
#else
    {
        float s = 0.0f;
        for (int i = lane; i < 4096; i += 32) s += img[i];
        for (int off = 16; off > 0; off >>= 1) s += __shfl_down(s, off, 32);
        total = __shfl(s, 0, 32);
    }
#endif

    // ---- per-pixel greedy distance-ordered mass consumption ---------------
    const int p = rb * 256 + (int)threadIdx.x;   // pixel index in image
    const int y = p >> 6;
    const int x = p & 63;
    const float m_target = 0.01f * total;

    __builtin_prefetch(table, 0, 1);             // global_prefetch_b8
    __builtin_prefetch(table + 16, 0, 1);

    float cum = 0.0f;     // cumulative *full* weights (matches reference cumsum)
    float wsum = 0.0f;    // sum of w_effective * d2
    for (int k = 0; k < NOFFS; ++k) {
        int e  = table[k];                       // uniform per wave -> L2 hit
        int d2 = e >> 16;
        int ny = y + (((e >> 8) & 0xff) - 63);
        int nx = x + ((e & 0xff) - 63);
        if ((unsigned)ny < 64u && (unsigned)nx < 64u) {
            float w    = img[ny * 64 + nx];
            float take = fmaxf(fminf(w, m_target - cum), 0.0f);
            wsum += take * (float)d2;
            cum  += w;
            if (cum >= m_target) break;          // all later takes are 0
        }
    }

    float dtm = (total == 0.0f) ? 0.0f : sqrtf(wsum / m_target);
    out[b * 4096 + p] = dtm;
}

// ------------------------------- launcher ----------------------------------

extern "C" void kernel_launch(void* const* d_in, const int* in_sizes, int n_in,
                              void* d_out, int out_size, void* d_ws, size_t ws_size,
                              hipStream_t stream) {
    (void)in_sizes; (void)n_in; (void)out_size; (void)ws_size;
    const float* images = (const float*)d_in[0];
    float* out = (float*)d_out;

    int* hist  = (int*)d_ws;        // NDIST ints
    int* table = hist + NDIST;      // NOFFS ints

    dtm_hist_init<<<(NDIST + 255) / 256, 256, 0, stream>>>(hist);
    dtm_hist_count<<<(NOFFS + 255) / 256, 256, 0, stream>>>(hist);
    dtm_scan_scatter<<<1, 32, 0, stream>>>(hist, table);
    dtm_main<<<64, 256, 0, stream>>>(images, out, table);
}